// SAGE_53197464928927
// MI455X (gfx1250) — compile-verified
//
#include <hip/hip_runtime.h>

// GraphSAGE forward on gfx1250 (MI455X), wave32 + V_WMMA_F32_16X16X4_F32.
//
// Pipeline:
//   invdeg = 1/max(degree(dst),1)
//   h = relu(x @ lin0_w + lin0_b)                          (WMMA GEMM, K=256,M=128)
//   3x: agg = scatter_add(h[src] -> dst)                    (streaming atomics)
//       h   = relu((agg*invdeg) @ Wl + bl + h @ Wr)         (fused 2-pass WMMA GEMM)
//   out = h @ lin1_w + lin1_b                               (WMMA GEMM, K=128,M=64)

typedef __attribute__((ext_vector_type(2))) float f32x2;
typedef __attribute__((ext_vector_type(8))) float f32x8;

#define TB 256  // 8 waves of 32

// ---------------------------------------------------------------------------
// Small streaming kernels
// ---------------------------------------------------------------------------
__global__ void zero_kernel(float* __restrict__ p, int n) {
    int i = blockIdx.x * blockDim.x + threadIdx.x;
    if (i < n) p[i] = 0.0f;
}

__global__ void deg_kernel(const int* __restrict__ dst, float* __restrict__ deg, int E) {
    int e = blockIdx.x * blockDim.x + threadIdx.x;
    if (e < E) atomicAdd(&deg[dst[e]], 1.0f);
}

__global__ void invdeg_kernel(float* __restrict__ deg, int n) {
    int i = blockIdx.x * blockDim.x + threadIdx.x;
    if (i < n) deg[i] = 1.0f / fmaxf(deg[i], 1.0f);
}

// One (edge, feature) pair per lane: 128 consecutive lanes stream one edge's
// feature row (fully coalesced 512B read) and scatter with f32 atomics.
__global__ void scatter_kernel(const float* __restrict__ h,
                               const int* __restrict__ src,
                               const int* __restrict__ dst,
                               float* __restrict__ agg, int E) {
    long long tid = (long long)blockIdx.x * blockDim.x + threadIdx.x;
    int e = (int)(tid >> 7);   // H = 128
    int f = (int)(tid & 127);
    if (e < E) {
        int s = src[e];
        int d = dst[e];
        atomicAdd(&agg[(size_t)d * 128 + f], h[(size_t)s * 128 + f]);
    }
}

// ---------------------------------------------------------------------------
// WMMA fp32 GEMM core.
// ISA layouts (wave32):
//   A 16x4 f32 : lane = M (lane&15), K = vgpr_j + 2*(lane>>4)   -> f32x2 load
//   B 4x16 f32 : lane = N (lane&15), K = vgpr_j + 2*(lane>>4)
//   C/D 16x16  : lane = N (lane&15), M = vgpr_r + 8*(lane>>4)
// Each wave computes a 16-row stripe over all M columns (M/16 v8f accums).
// W is staged through LDS in KC-row chunks with coalesced cooperative loads.
// ---------------------------------------------------------------------------
template <int K, int M, int KC>
__device__ __forceinline__ void gemm_pass(const float* __restrict__ A,
                                          const float* __restrict__ W,
                                          float ascale, int rowA,
                                          float* __restrict__ wlds,
                                          f32x8* acc) {
    const int tid    = threadIdx.x;
    const int lane   = tid & 31;
    const int laneLo = lane & 15;
    const int laneHi = lane >> 4;
    constexpr int NT = M / 16;

    for (int k0 = 0; k0 < K; k0 += KC) {
        __syncthreads();
        // Coalesced: W rows [k0, k0+KC) are contiguous in row-major W.
        for (int i = tid; i < KC * M; i += TB) wlds[i] = W[k0 * M + i];
        __syncthreads();
#pragma unroll
        for (int kk = 0; kk < KC; kk += 4) {
            f32x2 a = *(const f32x2*)(A + (size_t)rowA * K + (k0 + kk + 2 * laneHi));
            a.x *= ascale;
            a.y *= ascale;
#pragma unroll
            for (int t = 0; t < NT; ++t) {
                f32x2 b;
                b.x = wlds[(kk + 2 * laneHi + 0) * M + t * 16 + laneLo];
                b.y = wlds[(kk + 2 * laneHi + 1) * M + t * 16 + laneLo];
                acc[t] = __builtin_amdgcn_wmma_f32_16x16x4_f32(
                    /*neg_a=*/false, a, /*neg_b=*/false, b,
                    /*c_mod=*/(short)0, acc[t], /*reuse_a=*/false, /*reuse_b=*/false);
            }
        }
    }
}

template <int M, bool RELU>
__device__ __forceinline__ void epilogue(const f32x8* acc, const float* __restrict__ bias,
                                         float* __restrict__ C, int rowBase,
                                         int laneLo, int laneHi, int N) {
    constexpr int NT = M / 16;
#pragma unroll
    for (int t = 0; t < NT; ++t) {
        int col  = t * 16 + laneLo;
        float bv = bias[col];
#pragma unroll
        for (int r = 0; r < 8; ++r) {
            int row = rowBase + r + 8 * laneHi;
            if (row < N) {
                float v = acc[t][r] + bv;
                if (RELU) v = fmaxf(v, 0.0f);
                C[(size_t)row * M + col] = v;
            }
        }
    }
}

template <int K, int M, bool RELU>
__global__ void __launch_bounds__(TB) gemm_kernel(const float* __restrict__ A,
                                                  const float* __restrict__ W,
                                                  const float* __restrict__ bias,
                                                  float* __restrict__ C, int N) {
    constexpr int KC = 32;
    constexpr int NT = M / 16;
    __shared__ float wlds[KC * M];

    const int wave    = threadIdx.x >> 5;
    const int lane    = threadIdx.x & 31;
    const int laneLo  = lane & 15;
    const int laneHi  = lane >> 4;
    const int rowBase = blockIdx.x * 128 + wave * 16;
    int rowc = rowBase + laneLo;
    if (rowc > N - 1) rowc = N - 1;  // clamp loads; stores are predicated

    f32x8 acc[NT] = {};
    gemm_pass<K, M, KC>(A, W, 1.0f, rowc, wlds, acc);
    epilogue<M, RELU>(acc, bias, C, rowBase, laneLo, laneHi, N);
}

// Fused SAGE layer: h_out = relu((agg*invdeg) @ Wl + bl + h @ Wr)
// Two K-passes accumulate into the same WMMA accumulators (avoids a 25 MB
// intermediate round-trip per layer).
__global__ void __launch_bounds__(TB) sage_update_kernel(
    const float* __restrict__ agg, const float* __restrict__ invdeg,
    const float* __restrict__ h, const float* __restrict__ Wl,
    const float* __restrict__ bl, const float* __restrict__ Wr,
    float* __restrict__ hout, int N) {
    constexpr int K = 128, M = 128, KC = 32, NT = M / 16;
    __shared__ float wlds[KC * M];

    const int wave    = threadIdx.x >> 5;
    const int lane    = threadIdx.x & 31;
    const int laneLo  = lane & 15;
    const int laneHi  = lane >> 4;
    const int rowBase = blockIdx.x * 128 + wave * 16;
    int rowc = rowBase + laneLo;
    if (rowc > N - 1) rowc = N - 1;

    const float inv = invdeg[rowc];

    f32x8 acc[NT] = {};
    gemm_pass<K, M, KC>(agg, Wl, inv, rowc, wlds, acc);   // mean-aggregated neighbors
    gemm_pass<K, M, KC>(h, Wr, 1.0f, rowc, wlds, acc);    // root (self) features
    epilogue<M, true>(acc, bl, hout, rowBase, laneLo, laneHi, N);
}

// ---------------------------------------------------------------------------
// Host-side launch
// ---------------------------------------------------------------------------
extern "C" void kernel_launch(void* const* d_in, const int* in_sizes, int n_in,
                              void* d_out, int out_size, void* d_ws, size_t ws_size,
                              hipStream_t stream) {
    const float* x       = (const float*)d_in[0];
    const int*   edge    = (const int*)d_in[1];
    const float* lin0_w  = (const float*)d_in[2];
    const float* lin0_b  = (const float*)d_in[3];
    const float* conv_wl = (const float*)d_in[4];
    const float* conv_bl = (const float*)d_in[5];
    const float* conv_wr = (const float*)d_in[6];
    const float* lin1_w  = (const float*)d_in[7];
    const float* lin1_b  = (const float*)d_in[8];

    const int IN = 256, H = 128, L = 3;
    const int N = in_sizes[0] / IN;
    const int E = in_sizes[1] / 2;

    const int* src = edge;       // edge_index[0]
    const int* dst = edge + E;   // edge_index[1]

    // Workspace carve-up (needs ~77 MB): h0, h1, agg [N*H f32], invdeg [N f32]
    char*  ws     = (char*)d_ws;
    float* h0     = (float*)ws;  ws += (size_t)N * H * sizeof(float);
    float* h1     = (float*)ws;  ws += (size_t)N * H * sizeof(float);
    float* agg    = (float*)ws;  ws += (size_t)N * H * sizeof(float);
    float* invdeg = (float*)ws;  ws += (size_t)N * sizeof(float);

    // 1) degree -> 1/max(deg,1)
    zero_kernel<<<(N + 255) / 256, 256, 0, stream>>>(invdeg, N);
    deg_kernel<<<(E + 255) / 256, 256, 0, stream>>>(dst, invdeg, E);
    invdeg_kernel<<<(N + 255) / 256, 256, 0, stream>>>(invdeg, N);

    const int gblocks = (N + 127) / 128;

    // 2) h = relu(x @ lin0_w + lin0_b)
    gemm_kernel<256, 128, true><<<gblocks, TB, 0, stream>>>(x, lin0_w, lin0_b, h0, N);

    // 3) SAGE layers
    float* hcur  = h0;
    float* hnext = h1;
    const long long scat_threads = (long long)E * H;
    const int scat_blocks = (int)((scat_threads + 255) / 256);
    for (int i = 0; i < L; ++i) {
        zero_kernel<<<(N * H + 255) / 256, 256, 0, stream>>>(agg, N * H);
        scatter_kernel<<<scat_blocks, 256, 0, stream>>>(hcur, src, dst, agg, E);
        sage_update_kernel<<<gblocks, TB, 0, stream>>>(
            agg, invdeg, hcur,
            conv_wl + (size_t)i * H * H, conv_bl + (size_t)i * H,
            conv_wr + (size_t)i * H * H, hnext, N);
        float* t = hcur; hcur = hnext; hnext = t;
    }

    // 4) out = h @ lin1_w + lin1_b
    gemm_kernel<128, 64, false><<<gblocks, TB, 0, stream>>>(hcur, lin1_w, lin1_b,
                                                            (float*)d_out, N);
}